// PureHausdorff_59322088292633
// MI455X (gfx1250) — compile-verified
//
#include <hip/hip_runtime.h>

#define BB 8
#define HH 256
#define WW 256
#define HWSZ (HH*WW)
#define BHW (BB*HH*WW)
#define BIGD 512.0f              // H + W, matches reference cap
#define NELEM (BB*HH*WW)
#define NPART (BB*HH)            // 2048 per-row partial sums

typedef float v2f __attribute__((ext_vector_type(2)));
typedef float v8f __attribute__((ext_vector_type(8)));

// ---------------------------------------------------------------------------
// Workspace layout (floats unless noted):
//   [0, 4*BHW)          : g^2 planes, plane p: 0=x_fg 1=x_bg 2=y_fg 3=y_bg
//   [4*BHW, 4*BHW+NPART): per-row partial loss sums
//   then 16 ints        : has_fg flags (x: [0..7], y: [8..15])
// ---------------------------------------------------------------------------

__global__ void k_zero_flags(int* __restrict__ flags) {
    int i = threadIdx.x;
    if (i < 16) flags[i] = 0;
}

// One thread per (plane, batch, column): fused threshold + forward scan +
// backward scan + min + square, all in registers; g^2 written in place.
__global__ void k_scan(const float* __restrict__ x, const float* __restrict__ y,
                       float* __restrict__ g2, int* __restrict__ flags) {
    int idx = blockIdx.x * blockDim.x + threadIdx.x;   // 0 .. 4*B*W-1 = 8191
    int p   = idx >> 11;            // / (B*W) == /2048
    int rem = idx & 2047;
    int b   = rem >> 8;             // / W
    int w   = rem & 255;            // % W
    const float* in = (p < 2) ? x : y;
    int pol = p & 1;                // 0: fg mask, 1: bg mask (~fg)
    const float* col = in + (size_t)b * HWSZ + w;
    float* plane = g2 + (size_t)p * BHW + (size_t)b * HWSZ + w;

    // forward scan: d = m ? min(carry+1, BIG) : 0
    float carry = BIGD;
    int any_fg = 0;
    for (int h = 0; h < HH; ++h) {
        float v = col[(size_t)h * WW];
        if (h + 1 < HH)
            __builtin_prefetch(&col[(size_t)(h + 1) * WW], 0, 0); // global_prefetch_b8
        bool fg = (v > 0.5f);
        any_fg |= (int)fg;
        bool m = pol ? !fg : fg;
        float d = m ? fminf(carry + 1.0f, BIGD) : 0.0f;
        carry = d;
        plane[(size_t)h * WW] = d;      // fwd distance (0 iff mask False)
    }
    if (pol == 0 && any_fg)
        atomicOr(&flags[(p >> 1) * 8 + b], 1);   // idempotent -> deterministic

    // backward scan fused with min(fwd,bwd) and squaring, in place.
    // fwd value is 0 iff mask is False (fwd >= 1 on True pixels), so the
    // mask is recoverable without re-reading the input.
    carry = BIGD;
    for (int h = HH - 1; h >= 0; --h) {
        float f = plane[(size_t)h * WW];
        bool m = (f != 0.0f);
        float d = m ? fminf(carry + 1.0f, BIGD) : 0.0f;
        carry = d;
        float g = fminf(f, d);
        plane[(size_t)h * WW] = g * g;
    }
}

// One block per (b,h) row: brute-force lower envelope over W for all 4 planes
// (LDS broadcast of g^2 rows), then the fused loss term + deterministic
// tree reduction into one partial per row.
__global__ void k_row(const float* __restrict__ x, const float* __restrict__ y,
                      const float* __restrict__ g2, const int* __restrict__ flags,
                      float* __restrict__ partials) {
    __shared__ float sg[4][WW];
    __shared__ float red[WW];
    int bh = blockIdx.x;            // 0 .. B*H-1
    int b  = bh >> 8;
    int h  = bh & 255;
    int j  = threadIdx.x;           // column
    size_t rowoff = (size_t)b * HWSZ + (size_t)h * WW;

    #pragma unroll
    for (int p = 0; p < 4; ++p)
        sg[p][j] = g2[(size_t)p * BHW + rowoff + j];
    __syncthreads();

    float d0 = 3.4e38f, d1 = 3.4e38f, d2 = 3.4e38f, d3 = 3.4e38f;
    float fj = (float)j;
    #pragma unroll 4
    for (int jp = 0; jp < WW; ++jp) {
        float dj = fj - (float)jp;
        float q  = dj * dj;
        d0 = fminf(d0, sg[0][jp] + q);
        d1 = fminf(d1, sg[1][jp] + q);
        d2 = fminf(d2, sg[2][jp] + q);
        d3 = fminf(d3, sg[3][jp] + q);
    }
    float dtx = sqrtf(d0) + sqrtf(d1);   // f_dist + b_dist (pointwise)
    float dty = sqrtf(d2) + sqrtf(d3);
    if (!flags[b])     dtx = 0.0f;       // where(has_fg, ., 0)
    if (!flags[8 + b]) dty = 0.0f;

    float xv = x[rowoff + j];
    float yv = y[rowoff + j];
    float e  = xv - yv;
    red[j] = e * e * (dtx * dtx + dty * dty);
    __syncthreads();
    #pragma unroll
    for (int s = 128; s > 0; s >>= 1) {
        if (j < s) red[j] += red[j + s];
        __syncthreads();
    }
    if (j == 0) partials[bh] = red[0];
}

// Single-wave WMMA reduction of NPART=2048 partials:
// D = A x Ones + C with V_WMMA_F32_16X16X4_F32; each WMMA folds 64 f32
// into the 16 row-sums of the accumulator; 32 chained WMMAs total.
// C/D layout (32-bit 16x16): lane 0 holds D[0..7][0] in v[0..7],
// lane 16 holds D[8..15][0] -> total = lane0 sum + lane16 sum.
__global__ void k_reduce(const float* __restrict__ partials, float* __restrict__ out) {
    __shared__ float tail[2];
    int lane = threadIdx.x;         // 0..31, EXEC all ones around the WMMAs
    v8f acc = {0.f, 0.f, 0.f, 0.f, 0.f, 0.f, 0.f, 0.f};
    v2f ones; ones.x = 1.0f; ones.y = 1.0f;
    int row = lane & 15;
    int k0  = (lane >> 4) << 1;     // lanes 0-15: K=0,1 ; lanes 16-31: K=2,3
    for (int c = 0; c < NPART / 64; ++c) {
        int base = c * 64 + row * 4 + k0;
        v2f a; a.x = partials[base]; a.y = partials[base + 1];
        acc = __builtin_amdgcn_wmma_f32_16x16x4_f32(
                  /*neg_a=*/false, a, /*neg_b=*/false, ones,
                  /*c_mod=*/(short)0, acc, /*reuse_a=*/false, /*reuse_b=*/false);
    }
    float s = acc[0] + acc[1] + acc[2] + acc[3] + acc[4] + acc[5] + acc[6] + acc[7];
    if (lane == 0)  tail[0] = s;
    if (lane == 16) tail[1] = s;
    __syncthreads();
    if (lane == 0)
        out[0] = (tail[0] + tail[1]) * (1.0f / (float)NELEM);
}

extern "C" void kernel_launch(void* const* d_in, const int* in_sizes, int n_in,
                              void* d_out, int out_size, void* d_ws, size_t ws_size,
                              hipStream_t stream) {
    (void)in_sizes; (void)n_in; (void)out_size; (void)ws_size;
    const float* x = (const float*)d_in[0];
    const float* y = (const float*)d_in[1];
    float* out = (float*)d_out;

    float* g2       = (float*)d_ws;                 // 4*BHW floats (8 MB)
    float* partials = g2 + (size_t)4 * BHW;         // NPART floats
    int*   flags    = (int*)(partials + NPART);     // 16 ints

    k_zero_flags<<<1, 32, 0, stream>>>(flags);
    k_scan<<<(4 * BB * WW) / 256, 256, 0, stream>>>(x, y, g2, flags);
    k_row<<<BB * HH, 256, 0, stream>>>(x, y, g2, flags, partials);
    k_reduce<<<1, 32, 0, stream>>>(partials, out);
}